// ChannelVideoAttention_8486855376942
// MI455X (gfx1250) — compile-verified
//
#include <hip/hip_runtime.h>

typedef __attribute__((ext_vector_type(2))) float v2f;
typedef __attribute__((ext_vector_type(8))) float v8f;

#define NB     4
#define NS     16
#define NT     16
#define NDIM   512
#define NHW    1024
#define NHEADS 8
#define NW     32
#define NP     8                       // pixels per workgroup tile
#define NCOLS  (NT * NP)               // 128 columns = (t,p)
#define STG_STRIDE 68                  // 64 c' + 4 pad dwords (68%64==4 -> b64 conflict-free)
#define VP_STRIDE  129                 // 128 cols + 1 pad dword
#define VP_FLOATS  (NDIM * VP_STRIDE)  // 66048 floats
#define STG_OFF    VP_FLOATS
#define LDS_FLOATS (VP_FLOATS + NCOLS * STG_STRIDE)   // 74752 floats
#define LDS_BYTES  (LDS_FLOATS * 4)                   // 299008 B  (<320KB WGP LDS)

// ---------------------------------------------------------------------------
// Kernel 1: spatial mean pooling of q and k.  One wave32 per contiguous
// 1024-float row; 8 rows per 256-thread block.  Reads q+k (256MB) once.
// ---------------------------------------------------------------------------
__global__ __launch_bounds__(256) void pool_kernel(const float* __restrict__ q,
                                                   const float* __restrict__ k,
                                                   float* __restrict__ qm,
                                                   float* __restrict__ km) {
    const int row  = blockIdx.x * 8 + (threadIdx.x >> 5);   // [0, 2*32768)
    const int lane = threadIdx.x & 31;
    const float* src;
    float* dst;
    if (row < NB * NS * NDIM) {
        src = q + (size_t)row * NHW;
        dst = qm + row;
    } else {
        const int r2 = row - NB * NS * NDIM;
        src = k + (size_t)r2 * NHW;
        dst = km + r2;
    }
    float s = 0.0f;
#pragma unroll
    for (int i = 0; i < 8; ++i) {
        const float4 x = *reinterpret_cast<const float4*>(src + i * 128 + lane * 4);
        s += x.x + x.y + x.z + x.w;
    }
#pragma unroll
    for (int off = 16; off > 0; off >>= 1) s += __shfl_xor(s, off, 32);
    if (lane == 0) *dst = s * (1.0f / 1024.0f);
}

// ---------------------------------------------------------------------------
// Kernel 2: tiny QK projections + softmax -> attn[B][8][16][16].
// One block per batch.  qh gets the dh^-0.5 scale (dh=64 -> 1/8).
// ---------------------------------------------------------------------------
__global__ __launch_bounds__(256) void attn_kernel(const float* __restrict__ qm,
                                                   const float* __restrict__ km,
                                                   const float* __restrict__ Wq,
                                                   const float* __restrict__ Wk,
                                                   float* __restrict__ attn) {
    extern __shared__ float sm[];
    float* qh = sm;                 // [16][512]
    float* kh = sm + NS * NDIM;     // [16][512]
    const int b = blockIdx.x, tid = threadIdx.x;
    const float scale = 0.125f;

    for (int o = tid; o < 2 * NS * NDIM; o += 256) {
        const int which = o >> 13;                     // 0: qh, 1: kh
        const int idx = o & (NS * NDIM - 1);
        const int s = idx >> 9, c = idx & (NDIM - 1);
        const float* W = which ? Wk : Wq;
        const float* X = (which ? km : qm) + (size_t)b * NS * NDIM;
        float acc = 0.0f;
        for (int cc = 0; cc < NDIM; ++cc) acc += X[s * NDIM + cc] * W[c * NDIM + cc];
        (which ? kh : qh)[s * NDIM + c] = which ? acc : acc * scale;
    }
    __syncthreads();

    if (tid < NHEADS * NS) {
        const int hd = tid >> 4, s = tid & 15;
        float lg[NT];
        float mx = -1e30f;
        for (int t = 0; t < NT; ++t) {
            float a = 0.0f;
            for (int d = 0; d < 64; ++d)
                a += qh[s * NDIM + hd * 64 + d] * kh[t * NDIM + hd * 64 + d];
            lg[t] = a;
            mx = fmaxf(mx, a);
        }
        float den = 0.0f;
        for (int t = 0; t < NT; ++t) { lg[t] = __expf(lg[t] - mx); den += lg[t]; }
        const float inv = 1.0f / den;
        for (int t = 0; t < NT; ++t)
            attn[((size_t)(b * NHEADS + hd) * NS + s) * NT + t] = lg[t] * inv;
    }
}

// ---------------------------------------------------------------------------
// Kernel 3 (main): per (batch, 8-pixel tile):
//   Phase 1:  Vp[c, col=(t*8+p)] = sum_c' Wv[c,c'] * v[b,t,c',pix0+p]
//             M=512 N=128 K=512 via V_WMMA_F32_16X16X4_F32; Vp -> LDS.
//   Phase 2:  wave w == head w:  out[s,(c,p)] = A_hd(16x16 over t) @ Vp_hd.
// Spatial transpose of the reference handled at the final scatter-store
// (output fits in 192MB L2, which absorbs the 4B-granular writes).
// ---------------------------------------------------------------------------
__global__ __launch_bounds__(256) void cva_main(const float* __restrict__ v,
                                                const float* __restrict__ Wv,
                                                const float* __restrict__ attn,
                                                float* __restrict__ out) {
    extern __shared__ float lds[];
    float* vp    = lds;             // [512][VP_STRIDE]
    float* stage = lds + STG_OFF;   // [128 cols][STG_STRIDE], 64-wide c' chunk

    const int tid  = threadIdx.x;
    const int wave = tid >> 5;      // 0..7 == head id in phase 2
    const int lane = tid & 31;
    const int lm   = lane & 15;
    const int lh   = lane >> 4;

    const int blk  = blockIdx.x;                 // [0, NB*NHW/NP)
    const int b    = blk / (NHW / NP);
    const int pix0 = (blk % (NHW / NP)) * NP;    // memory pixel = h*32 + w
    const int hh   = pix0 / NW;                  // fixed v-space h for this tile
    const int ww0  = pix0 % NW;

    // ---------------- Phase 1 ----------------
    for (int mp = 0; mp < 2; ++mp) {             // 2 M-tile pairs -> 64 c rows/wave
        v8f acc[2][8];
#pragma unroll
        for (int m = 0; m < 2; ++m)
#pragma unroll
            for (int n = 0; n < 8; ++n) acc[m][n] = (v8f)0.0f;

        for (int kc = 0; kc < 8; ++kc) {         // K in 64-wide chunks
            __syncthreads();                     // previous chunk fully consumed
            // cooperative stage load: 64 c' x 16 t x 8 p (transposing to [col][c'])
#pragma unroll
            for (int i = 0; i < 4; ++i) {
                const int r  = tid + i * 256;    // 0..1023
                const int t  = r >> 6;
                const int cs = r & 63;
                const float* gp =
                    v + ((size_t)(b * NT + t) * NDIM + kc * 64 + cs) * NHW + pix0;
                const float4 x0 = *reinterpret_cast<const float4*>(gp);
                const float4 x1 = *reinterpret_cast<const float4*>(gp + 4);
                const float vals[8] = {x0.x, x0.y, x0.z, x0.w, x1.x, x1.y, x1.z, x1.w};
#pragma unroll
                for (int p = 0; p < 8; ++p)
                    stage[(t * NP + p) * STG_STRIDE + cs] = vals[p];
            }
            __syncthreads();

            for (int ks = 0; ks < 16; ++ks) {
                const int klocal = ks * 4 + 2 * lh;
                const int kglob  = kc * 64 + klocal;
                v2f a[2];
#pragma unroll
                for (int m = 0; m < 2; ++m) {
                    const int c = wave * 64 + (mp * 2 + m) * 16 + lm;
                    a[m] = *reinterpret_cast<const v2f*>(Wv + (size_t)c * NDIM + kglob);
                }
                v2f bf[8];
#pragma unroll
                for (int n = 0; n < 8; ++n) {
                    const int col = n * 16 + lm;
                    bf[n] = *reinterpret_cast<const v2f*>(stage + col * STG_STRIDE + klocal);
                }
#pragma unroll
                for (int m = 0; m < 2; ++m)
#pragma unroll
                    for (int n = 0; n < 8; ++n)
                        acc[m][n] = __builtin_amdgcn_wmma_f32_16x16x4_f32(
                            false, a[m], false, bf[n], (short)0, acc[m][n], false, false);
            }
        }
        // spill this M-tile pair's Vp into LDS (disjoint from stage)
#pragma unroll
        for (int m = 0; m < 2; ++m)
#pragma unroll
            for (int n = 0; n < 8; ++n)
#pragma unroll
                for (int e = 0; e < 8; ++e) {
                    const int c   = wave * 64 + (mp * 2 + m) * 16 + e + 8 * lh;
                    const int col = n * 16 + lm;
                    vp[c * VP_STRIDE + col] = acc[m][n][e];
                }
    }
    __syncthreads();

    // ---------------- Phase 2 ----------------
    // A2[m=s][k=t] = attn[b][head=wave][s][t]
    v2f a2[4];
    const float* Ab = attn + ((size_t)(b * NHEADS + wave) * NS + lm) * NT;
#pragma unroll
    for (int kk = 0; kk < 4; ++kk)
        a2[kk] = *reinterpret_cast<const v2f*>(Ab + kk * 4 + 2 * lh);

    const int n2      = /*per-lane column*/ 0;
    (void)n2;
    for (int nt = 0; nt < 32; ++nt) {            // N = 64 c_local x 8 p = 512
        const int nn = nt * 16 + lm;
        const int cl = nn >> 3;                  // c within head
        const int p  = nn & 7;
        const int c  = wave * 64 + cl;
        v8f o = (v8f)0.0f;
#pragma unroll
        for (int kk = 0; kk < 4; ++kk) {
            const int t0 = kk * 4 + 2 * lh;
            v2f bb;
            bb.x = vp[c * VP_STRIDE + (t0 + 0) * NP + p];
            bb.y = vp[c * VP_STRIDE + (t0 + 1) * NP + p];
            o = __builtin_amdgcn_wmma_f32_16x16x4_f32(false, a2[kk], false, bb,
                                                      (short)0, o, false, false);
        }
        // reference spatial transpose: v-pixel (hh, ww0+p) -> out[..., ww0+p, hh]
        const size_t part = (size_t)c * NHW + (size_t)(ww0 + p) * NW + hh;
#pragma unroll
        for (int e = 0; e < 8; ++e) {
            const int s = e + 8 * lh;
            out[(size_t)(b * NS + s) * (NDIM * NHW) + part] = o[e];
        }
    }
}

// ---------------------------------------------------------------------------
extern "C" void kernel_launch(void* const* d_in, const int* in_sizes, int n_in,
                              void* d_out, int out_size, void* d_ws, size_t ws_size,
                              hipStream_t stream) {
    const float* q  = (const float*)d_in[0];
    const float* k  = (const float*)d_in[1];
    const float* v  = (const float*)d_in[2];
    const float* Wq = (const float*)d_in[3];
    const float* Wk = (const float*)d_in[4];
    const float* Wv = (const float*)d_in[5];
    float* out = (float*)d_out;

    float* qm   = (float*)d_ws;                       // 32768 floats
    float* km   = qm + NB * NS * NDIM;                // 32768 floats
    float* attn = km + NB * NT * NDIM;                // 8192 floats

    // allow >64KB dynamic LDS (deterministic, idempotent)
    (void)hipFuncSetAttribute(reinterpret_cast<const void*>(&cva_main),
                              hipFuncAttributeMaxDynamicSharedMemorySize, LDS_BYTES);
    (void)hipFuncSetAttribute(reinterpret_cast<const void*>(&attn_kernel),
                              hipFuncAttributeMaxDynamicSharedMemorySize,
                              2 * NS * NDIM * 4);

    pool_kernel<<<(2 * NB * NS * NDIM) / 8, 256, 0, stream>>>(q, k, qm, km);
    attn_kernel<<<NB, 256, 2 * NS * NDIM * 4, stream>>>(qm, km, Wq, Wk, attn);
    cva_main<<<NB * (NHW / NP), 256, LDS_BYTES, stream>>>(v, Wv, attn, out);
}